// WeakNegativeLoss_1486058684920
// MI455X (gfx1250) — compile-verified
//
#include <hip/hip_runtime.h>

typedef __attribute__((ext_vector_type(2))) float v2f;
typedef __attribute__((ext_vector_type(8))) float v8f;

namespace {
constexpr int   kB = 256;
constexpr int   kL = 512;
constexpr int   kD = 64;
constexpr float kEpsLog = 1e-7f;
constexpr float kEpsCos = 1e-8f;
}

// One block per batch sample. 128 threads = 4 wave32.
// Computes per-batch partial:  -sum_l y*log(p+eps)  -  wnl_b * sum_l log(1-(p+eps))
__global__ __launch_bounds__(128)
void wnl_main_kernel(const float* __restrict__ emb,     // (B, L, D) fp32
                     const float* __restrict__ probs,   // (B, L)
                     const float* __restrict__ labels,  // (B, L)
                     float* __restrict__ ws)            // (B,) partials
{
    const int b    = blockIdx.x;
    const int tid  = threadIdx.x;
    const int lane = tid & 31;
    const int wave = tid >> 5;
    const int n    = lane & 15;   // key index within 16-wide tile
    const int g    = lane >> 4;   // lane-half (selects K sub-pair per ISA layout)

    __shared__ float s_max[4];
    __shared__ float s_pos[4];
    __shared__ float s_ln[4];

    const float* labrow = labels + (size_t)b * kL;
    const float* prow   = probs  + (size_t)b * kL;

    // ---------- A fragments: all 16 rows = q = emb[b, b, :] ----------
    // f32 A 16x4 layout: lanes 0-15 hold {K=4t, 4t+1}, lanes 16-31 hold {4t+2, 4t+3}.
    // Rows are all identical (broadcast q), so the fragment depends only on lane half.
    const float* qp = emb + ((size_t)b * kL + b) * kD;
    v2f a[16];
    float nq2l = 0.f;
    #pragma unroll
    for (int t = 0; t < 16; ++t) {
        a[t] = *(const v2f*)(qp + 4 * t + 2 * g);
        nq2l += a[t].x * a[t].x + a[t].y * a[t].y;
    }
    // lane half 0 summed K = {0,1} mod 4, half 1 summed {2,3} mod 4 -> combine.
    const float nq2 = nq2l + __shfl_xor(nq2l, 16, 32);
    const float nq  = sqrtf(nq2);

    // ---------- masked row max over the 32 key tiles (8 tiles per wave) ----------
    float m = 0.0f;   // all masked values are >= 0
    for (int tc = wave; tc < kL / 16; tc += 4) {   // wave-uniform: EXEC stays all-1s
        const int k0 = tc * 16;
        const float* kp = emb + ((size_t)b * kL + k0 + n) * kD;

        // f32 B 4x16 layout: VGPR0 holds row K=(2g), VGPR1 row K=(2g+1), cols = lanes.
        v2f bf[16];
        float nk2l = 0.f;
        #pragma unroll
        for (int t = 0; t < 16; ++t) {
            bf[t] = *(const v2f*)(kp + 4 * t + 2 * g);
            nk2l += bf[t].x * bf[t].x + bf[t].y * bf[t].y;
        }
        const float nk2 = nk2l + __shfl_xor(nk2l, 16, 32);  // lane l and l+16 share key n

        // Chain 16 exact-fp32 WMMAs over K=64: D = A(16x4) x B(4x16) + C
        v8f c = {};
        #pragma unroll
        for (int t = 0; t < 16; ++t) {
            c = __builtin_amdgcn_wmma_f32_16x16x4_f32(
                    /*neg_a=*/false, a[t], /*neg_b=*/false, bf[t],
                    /*c_mod=*/(short)0, c, /*reuse_a=*/false, /*reuse_b=*/false);
        }
        // All A rows equal q => every D row is the dot row; c[0] lanes 0-15 are
        // M=0,N=lane and lanes 16-31 are M=8,N=lane-16 (identical rows).
        const float dot   = c[0];
        const float nk    = sqrtf(nk2);
        const float denom = fmaxf(nq * nk, kEpsCos);
        const float cosv  = dot / denom;
        const float beta  = fmaxf(cosv, 0.0f);       // relu(cos - theta), theta = 0
        const float msk   = beta * labrow[k0 + n];
        m = fmaxf(m, msk);
    }
    #pragma unroll
    for (int s = 16; s > 0; s >>= 1) m = fmaxf(m, __shfl_xor(m, s, 32));
    if (lane == 0) s_max[wave] = m;

    // ---------- log sums over the probs row ----------
    float pos = 0.f, ln = 0.f;
    for (int l = tid; l < kL; l += 128) {
        const float p = prow[l] + kEpsLog;
        pos += labrow[l] * logf(p);
        ln  += logf(1.0f - p);
    }
    #pragma unroll
    for (int s = 16; s > 0; s >>= 1) {
        pos += __shfl_xor(pos, s, 32);
        ln  += __shfl_xor(ln,  s, 32);
    }
    if (lane == 0) { s_pos[wave] = pos; s_ln[wave] = ln; }

    __syncthreads();
    if (tid == 0) {
        const float rmax = fmaxf(fmaxf(s_max[0], s_max[1]), fmaxf(s_max[2], s_max[3]));
        const float psum = (s_pos[0] + s_pos[1]) + (s_pos[2] + s_pos[3]);
        const float lsum = (s_ln[0]  + s_ln[1])  + (s_ln[2]  + s_ln[3]);
        const float wnl  = (labrow[b] == 0.0f) ? rmax : 0.0f;  // diagonal pick (L >= B)
        ws[b] = -psum - wnl * lsum;
    }
}

// Deterministic final reduction of the 256 per-batch partials.
__global__ __launch_bounds__(256)
void wnl_reduce_kernel(const float* __restrict__ ws, float* __restrict__ out)
{
    __shared__ float s[kB];
    const int t = threadIdx.x;
    s[t] = ws[t];
    __syncthreads();
    for (int st = kB / 2; st > 0; st >>= 1) {
        if (t < st) s[t] += s[t + st];
        __syncthreads();
    }
    if (t == 0) out[0] = s[0] / (float)kB;
}

extern "C" void kernel_launch(void* const* d_in, const int* in_sizes, int n_in,
                              void* d_out, int out_size, void* d_ws, size_t ws_size,
                              hipStream_t stream)
{
    const float* emb    = (const float*)d_in[0];  // (256, 512, 64)
    const float* probs  = (const float*)d_in[1];  // (256, 512)
    const float* labels = (const float*)d_in[2];  // (256, 512)
    float*       out    = (float*)d_out;          // scalar
    float*       ws     = (float*)d_ws;           // >= 256 floats

    wnl_main_kernel<<<kB, 128, 0, stream>>>(emb, probs, labels, ws);
    wnl_reduce_kernel<<<1, kB, 0, stream>>>(ws, out);
}